// MultiHeadSelectiveAttention_67242007986891
// MI455X (gfx1250) — compile-verified
//
#include <hip/hip_runtime.h>

typedef __attribute__((ext_vector_type(2))) float v2f;
typedef __attribute__((ext_vector_type(4))) float v4f;
typedef __attribute__((ext_vector_type(8))) float v8f;

#define NB   8
#define LSEQ 4096
#define DEMB 1024
#define NH   16
#define DH   64

// D[16x16] += A[16x4] * B[4x16], all fp32 (full precision, matrix pipe).
__device__ __forceinline__ v8f wmma_f32(v2f a, v2f b, v8f c) {
  return __builtin_amdgcn_wmma_f32_16x16x4_f32(false, a, false, b, (short)0, c,
                                               false, false);
}

// ---------------- K1: V[b,l,h] = x[b,l,:]@Wv[:,h] + bv[h]; sv[b,h] += sum_l V
__global__ __launch_bounds__(128) void k_vproj(const float* __restrict__ x,
                                               const float* __restrict__ Wv,
                                               const float* __restrict__ bv,
                                               float* __restrict__ V,
                                               float* __restrict__ sv) {
  const int lane = threadIdx.x & 31;
  const int gid  = blockIdx.x * 4 + (threadIdx.x >> 5);  // 0..2047 row tiles
  const int m0   = gid << 4;                             // global row base
  const int n    = lane & 15;                            // head / A-row
  const int kh   = lane >> 4;                            // K half (0/1)

  v8f acc = {};
  const float* aptr = x + (size_t)(m0 + n) * DEMB + kh * 2;
#pragma unroll 4
  for (int k0 = 0; k0 < DEMB; k0 += 4) {
    v2f a = *(const v2f*)(aptr + k0);
    v2f b;
    b[0] = Wv[(k0 + kh * 2 + 0) * NH + n];
    b[1] = Wv[(k0 + kh * 2 + 1) * NH + n];
    acc = wmma_f32(a, b, acc);
  }
  const float bvn = bv[n];
  float colsum = 0.f;
#pragma unroll
  for (int r = 0; r < 8; ++r) {
    float vv = acc[r] + bvn;                       // D: M = r + kh*8, N = n
    V[(size_t)(m0 + kh * 8 + r) * NH + n] = vv;
    colsum += vv;
  }
  atomicAdd(&sv[(m0 >> 12) * NH + n], colsum);
}

// ---------------- K2: XV[b,h,c] = sum_l V[b,l,h] * x[b,l,c]  (K-split x4)
__global__ __launch_bounds__(128) void k_xv(const float* __restrict__ x,
                                            const float* __restrict__ V,
                                            float* __restrict__ XV) {
  const int lane = threadIdx.x & 31;
  const int gid  = blockIdx.x * 4 + (threadIdx.x >> 5);  // 0..2047
  const int b_   = gid >> 8;
  const int rem  = gid & 255;
  const int nt   = rem >> 2;   // 64 column tiles of D
  const int kc   = rem & 3;    // 4 K chunks of 1024 rows
  const int n    = lane & 15;
  const int kh   = lane >> 4;
  const int c0   = nt * 16;

  const float* Xb = x + (size_t)b_ * LSEQ * DEMB;
  const float* Vb = V + (size_t)b_ * LSEQ * NH;
  v8f acc = {};
  const int kbeg = kc * 1024;
#pragma unroll 4
  for (int k = kbeg; k < kbeg + 1024; k += 4) {
    const int l = k + kh * 2;
    v2f a;                               // A[M=h,K=l] = V[l,h]
    a[0] = Vb[(size_t)(l + 0) * NH + n];
    a[1] = Vb[(size_t)(l + 1) * NH + n];
    v2f b;                               // B[K=l,N=c] = x[l,c]
    b[0] = Xb[(size_t)(l + 0) * DEMB + c0 + n];
    b[1] = Xb[(size_t)(l + 1) * DEMB + c0 + n];
    acc = wmma_f32(a, b, acc);
  }
#pragma unroll
  for (int r = 0; r < 8; ++r) {
    const int h = kh * 8 + r;
    atomicAdd(&XV[((size_t)b_ * NH + h) * DEMB + c0 + n], acc[r]);
  }
}

// ---------------- K3 (tiny): ktv = XV@Wk_h + bk_h*sv ; wz = Wq_h@ktv ; zb = bq_h.ktv
__global__ __launch_bounds__(256) void k_ktv_wz(const float* __restrict__ XV,
                                                const float* __restrict__ Wk,
                                                const float* __restrict__ bk,
                                                const float* __restrict__ Wq,
                                                const float* __restrict__ bq,
                                                const float* __restrict__ sv,
                                                float* __restrict__ WZ,
                                                float* __restrict__ zb) {
  __shared__ float ktv[DH];
  __shared__ float red[256];
  const int bh = blockIdx.x;             // 0..127 = (b,h)
  const int h  = bh & 15;
  const float* xv = XV + (size_t)bh * DEMB;

  const int d = threadIdx.x >> 2, q = threadIdx.x & 3;  // 4 threads per d
  float s = 0.f;
  const int cbeg = q * 256;
  for (int c = cbeg; c < cbeg + 256; ++c)
    s += xv[c] * Wk[(size_t)c * DEMB + h * DH + d];
  red[threadIdx.x] = s;
  __syncthreads();
  if (q == 0) {
    float t = red[threadIdx.x] + red[threadIdx.x + 1] + red[threadIdx.x + 2] +
              red[threadIdx.x + 3];
    ktv[d] = t + bk[h * DH + d] * sv[bh];
  }
  __syncthreads();
  if (threadIdx.x == 0) {
    float z = 0.f;
    for (int dd = 0; dd < DH; ++dd) z += bq[h * DH + dd] * ktv[dd];
    zb[bh] = z;
  }
#pragma unroll 1
  for (int cc = 0; cc < 4; ++cc) {
    const int c = threadIdx.x * 4 + cc;
    float s2 = 0.f;
    for (int dd = 0; dd < DH; ++dd)
      s2 += Wq[(size_t)c * DEMB + h * DH + dd] * ktv[dd];
    WZ[(size_t)bh * DEMB + c] = s2;
  }
}

// ---------------- K4: out[b,h,l] = mask * sigmoid((x[b,l,:].wz[b,h,:] + zb)/8)
__global__ __launch_bounds__(128) void k_out(const float* __restrict__ x,
                                             const float* __restrict__ WZ,
                                             const float* __restrict__ zb,
                                             const unsigned char* __restrict__ mask,
                                             float* __restrict__ out) {
  const int lane = threadIdx.x & 31;
  const int gid  = blockIdx.x * 4 + (threadIdx.x >> 5);  // 0..2047
  const int b_   = gid >> 8;
  const int m0   = (gid & 255) << 4;   // row (l) base within batch
  const int n    = lane & 15;
  const int kh   = lane >> 4;

  const float* Xb  = x + (size_t)b_ * LSEQ * DEMB;
  const float* ar  = Xb + (size_t)(m0 + n) * DEMB + kh * 2;
  const float* wzr = WZ + ((size_t)b_ * NH + n) * DEMB + kh * 2;  // row h=n
  v8f acc = {};
#pragma unroll 4
  for (int k0 = 0; k0 < DEMB; k0 += 4) {
    v2f a = *(const v2f*)(ar + k0);
    v2f b = *(const v2f*)(wzr + k0);
    acc = wmma_f32(a, b, acc);
  }
  const float zbv   = zb[b_ * NH + n];
  const int   lbase = m0 + kh * 8;
  v4f o0, o1;
#pragma unroll
  for (int r = 0; r < 8; ++r) {
    const int l = lbase + r;
    float z = (acc[r] + zbv) * 0.125f;        // 1/sqrt(64)
    float p = 1.f / (1.f + __expf(-z));
    p = mask[b_ * LSEQ + l] ? p : 0.f;
    if (r < 4) o0[r] = p; else o1[r - 4] = p;
  }
  float* op = out + ((size_t)b_ * NH + n) * LSEQ + lbase;
  *(v4f*)op       = o0;
  *(v4f*)(op + 4) = o1;
}

extern "C" void kernel_launch(void* const* d_in, const int* in_sizes, int n_in,
                              void* d_out, int out_size, void* d_ws, size_t ws_size,
                              hipStream_t stream) {
  (void)in_sizes; (void)n_in; (void)out_size; (void)ws_size;
  const float* x  = (const float*)d_in[0];
  const unsigned char* mask = (const unsigned char*)d_in[1];
  const float* Wq = (const float*)d_in[2];
  const float* bq = (const float*)d_in[3];
  const float* Wk = (const float*)d_in[4];
  const float* bk = (const float*)d_in[5];
  const float* Wv = (const float*)d_in[6];
  const float* bv = (const float*)d_in[7];
  float* out = (float*)d_out;

  // V (B*L*NH floats == out_size) lives in d_out until K2 is done; K4 rewrites out.
  float* V  = out;
  float* XV = (float*)d_ws;                 // 8*16*1024
  float* WZ = XV + NB * NH * DEMB;          // 8*16*1024
  float* sv = WZ + NB * NH * DEMB;          // 128
  float* zb = sv + NB * NH;                 // 128

  hipMemsetAsync(d_ws, 0,
                 (size_t)(2 * NB * NH * DEMB + 2 * NB * NH) * sizeof(float),
                 stream);
  k_vproj <<<512, 128, 0, stream>>>(x, Wv, bv, V, sv);
  k_xv    <<<512, 128, 0, stream>>>(x, V, XV);
  k_ktv_wz<<<128, 256, 0, stream>>>(XV, Wk, bk, Wq, bq, sv, WZ, zb);
  k_out   <<<512, 128, 0, stream>>>(x, WZ, zb, mask, out);
}